// SubsamplingLayer_18674517803139
// MI455X (gfx1250) — compile-verified
//
#include <hip/hip_runtime.h>
#include <stdint.h>

#define B_ROWS   1024
#define D_COLS   65536
#define NTHREADS 1024      // 32 waves of wave32: fills one WGP
#define HBUCKETS 65536     // 2^16 radix buckets -> 256KB LDS (fits 320KB WGP LDS)
#define CHUNK    64        // buckets owned per thread
#define VEC_PER_THREAD ((D_COLS / 4) / NTHREADS)   // 16 float4 loads per thread per pass

// Order-preserving float->uint32 transform: a < b (float) <=> ord(a) < ord(b)
__device__ __forceinline__ uint32_t ord(float x) {
    uint32_t b = __float_as_uint(x);
    return (b & 0x80000000u) ? ~b : (b | 0x80000000u);
}

// ---------------------------------------------------------------------------
// Kernel 1: one workgroup per row; exact k-th largest via two 16-bit radix
// histogram passes held entirely in LDS (256KB histogram + 4KB scan array).
// Writes the row's k-th largest value (in ord-space) to rowk[row].
// ---------------------------------------------------------------------------
__global__ __launch_bounds__(NTHREADS, 1)
void topk_row_select(const float* __restrict__ in,
                     const int*   __restrict__ kptr,
                     uint32_t*    __restrict__ rowk)
{
    __shared__ uint32_t hist[HBUCKETS];   // 256 KB
    __shared__ uint32_t chunk[NTHREADS];  // 4 KB (suffix-scan workspace)
    __shared__ uint32_t sel[2];           // [0]=top16 bucket, [1]=rank within bucket

    const int tid = threadIdx.x;
    const int row = blockIdx.x;
    const uint32_t K = (uint32_t)(*kptr);
    const float4* rp = (const float4*)(in + (size_t)row * D_COLS);

    // ---------------- pass A: histogram of top 16 bits ----------------
    for (int i = tid; i < HBUCKETS; i += NTHREADS) hist[i] = 0u;
    __syncthreads();

    #pragma unroll 4
    for (int i = 0; i < VEC_PER_THREAD; ++i) {
        float4 v = rp[tid + i * NTHREADS];
        atomicAdd(&hist[ord(v.x) >> 16], 1u);
        atomicAdd(&hist[ord(v.y) >> 16], 1u);
        atomicAdd(&hist[ord(v.z) >> 16], 1u);
        atomicAdd(&hist[ord(v.w) >> 16], 1u);
    }
    __syncthreads();

    // per-thread chunk sums, then parallel inclusive SUFFIX sum (Hillis-Steele)
    uint32_t myc = 0;
    #pragma unroll
    for (int j = 0; j < CHUNK; ++j) myc += hist[tid * CHUNK + j];
    chunk[tid] = myc;
    __syncthreads();
    for (int off = 1; off < NTHREADS; off <<= 1) {
        uint32_t v = chunk[tid];
        uint32_t a = (tid + off < NTHREADS) ? chunk[tid + off] : 0u;
        __syncthreads();
        chunk[tid] = v + a;
        __syncthreads();
    }
    {   // exactly one thread owns the crossing point
        uint32_t incl = chunk[tid];
        uint32_t excl = incl - myc;
        if (excl < K && K <= incl) {
            uint32_t need = K - excl;
            int b = tid * CHUNK + CHUNK - 1;
            for (;; --b) {            // <=64 iterations, terminates inside chunk
                uint32_t h = hist[b];
                if (h >= need) break;
                need -= h;
            }
            sel[0] = (uint32_t)b;
            sel[1] = need;
        }
    }
    __syncthreads();
    const uint32_t b16 = sel[0];
    const uint32_t K2  = sel[1];

    // ---------------- pass B: low 16 bits within winning bucket ----------------
    for (int i = tid; i < HBUCKETS; i += NTHREADS) hist[i] = 0u;
    __syncthreads();

    #pragma unroll 4
    for (int i = 0; i < VEC_PER_THREAD; ++i) {
        float4 v = rp[tid + i * NTHREADS];   // re-read: row resident in L2
        uint32_t u;
        u = ord(v.x); if ((u >> 16) == b16) atomicAdd(&hist[u & 0xFFFFu], 1u);
        u = ord(v.y); if ((u >> 16) == b16) atomicAdd(&hist[u & 0xFFFFu], 1u);
        u = ord(v.z); if ((u >> 16) == b16) atomicAdd(&hist[u & 0xFFFFu], 1u);
        u = ord(v.w); if ((u >> 16) == b16) atomicAdd(&hist[u & 0xFFFFu], 1u);
    }
    __syncthreads();

    myc = 0;
    #pragma unroll
    for (int j = 0; j < CHUNK; ++j) myc += hist[tid * CHUNK + j];
    chunk[tid] = myc;
    __syncthreads();
    for (int off = 1; off < NTHREADS; off <<= 1) {
        uint32_t v = chunk[tid];
        uint32_t a = (tid + off < NTHREADS) ? chunk[tid + off] : 0u;
        __syncthreads();
        chunk[tid] = v + a;
        __syncthreads();
    }
    {
        uint32_t incl = chunk[tid];
        uint32_t excl = incl - myc;
        if (excl < K2 && K2 <= incl) {
            uint32_t need = K2 - excl;
            int b = tid * CHUNK + CHUNK - 1;
            for (;; --b) {
                uint32_t h = hist[b];
                if (h >= need) break;
                need -= h;
            }
            rowk[row] = (b16 << 16) | (uint32_t)b;   // exact k-th largest, ord-space
        }
    }
}

// ---------------------------------------------------------------------------
// Kernel 2: global min over the 1024 per-row keys; convert back to float.
// ---------------------------------------------------------------------------
__global__ __launch_bounds__(B_ROWS)
void reduce_threshold(const uint32_t* __restrict__ rowk, float* __restrict__ thr)
{
    __shared__ uint32_t sm[B_ROWS];
    const int tid = threadIdx.x;
    sm[tid] = rowk[tid];
    __syncthreads();
    for (int s = B_ROWS / 2; s > 0; s >>= 1) {
        if (tid < s) {
            uint32_t a = sm[tid], b = sm[tid + s];
            sm[tid] = (a < b) ? a : b;
        }
        __syncthreads();
    }
    if (tid == 0) {
        uint32_t u  = sm[0];
        uint32_t fb = (u & 0x80000000u) ? (u ^ 0x80000000u) : ~u;  // inverse ord()
        thr[0] = __uint_as_float(fb);
    }
}

// ---------------------------------------------------------------------------
// Kernel 3: streaming masked copy, b128 loads/stores + prefetch ahead.
// Reproduces (x>=t ? 1.0 : 0.0) * x exactly (incl. -0.0 cases).
// ---------------------------------------------------------------------------
__global__ __launch_bounds__(256)
void apply_threshold(const float* __restrict__ in,
                     const float* __restrict__ thr,
                     float*       __restrict__ out,
                     int n4)
{
    const float t = thr[0];
    const float4* ip = (const float4*)in;
    float4*       op = (float4*)out;
    const int stride = gridDim.x * blockDim.x;
    for (int i = blockIdx.x * blockDim.x + threadIdx.x; i < n4; i += stride) {
        if (i + stride < n4) __builtin_prefetch(&ip[i + stride], 0, 0);  // global_prefetch_b8
        float4 v = ip[i];
        float4 r;
        r.x = (v.x >= t ? 1.0f : 0.0f) * v.x;
        r.y = (v.y >= t ? 1.0f : 0.0f) * v.y;
        r.z = (v.z >= t ? 1.0f : 0.0f) * v.z;
        r.w = (v.w >= t ? 1.0f : 0.0f) * v.w;
        op[i] = r;
    }
}

extern "C" void kernel_launch(void* const* d_in, const int* in_sizes, int n_in,
                              void* d_out, int out_size, void* d_ws, size_t ws_size,
                              hipStream_t stream)
{
    const float* in   = (const float*)d_in[0];
    const int*   kptr = (const int*)d_in[1];
    float*       out  = (float*)d_out;

    uint32_t* rowk = (uint32_t*)d_ws;                                 // 1024 * u32
    float*    thr  = (float*)((char*)d_ws + B_ROWS * sizeof(uint32_t));

    hipLaunchKernelGGL(topk_row_select, dim3(B_ROWS), dim3(NTHREADS), 0, stream,
                       in, kptr, rowk);
    hipLaunchKernelGGL(reduce_threshold, dim3(1), dim3(B_ROWS), 0, stream,
                       rowk, thr);

    const int n4 = (B_ROWS * D_COLS) / 4;   // 16,777,216 float4s
    hipLaunchKernelGGL(apply_threshold, dim3(4096), dim3(256), 0, stream,
                       in, thr, out, n4);
}